// GraphSAGEModel_79869211837064
// MI455X (gfx1250) — compile-verified
//
#include <hip/hip_runtime.h>
#include <hip/hip_bf16.h>

typedef float v2f __attribute__((ext_vector_type(2)));
typedef float v8f __attribute__((ext_vector_type(8)));

#define GN 100000
#define GE 1600000

__global__ void fill_zero_kernel(float* __restrict__ p, long long n) {
    long long i = (long long)blockIdx.x * blockDim.x + threadIdx.x;
    if (i < n) p[i] = 0.0f;
}

__global__ void count_kernel(const long long* __restrict__ dst, float* __restrict__ cnt, int E) {
    int i = blockIdx.x * blockDim.x + threadIdx.x;
    if (i < E) {
        long long d = dst[i];
        __hip_atomic_fetch_add(&cnt[d], 1.0f, __ATOMIC_RELAXED, __HIP_MEMORY_SCOPE_AGENT);
    }
}

__global__ void inv_kernel(float* __restrict__ c, int n) {
    int i = blockIdx.x * blockDim.x + threadIdx.x;
    if (i < n) c[i] = 1.0f / fmaxf(c[i], 1.0f);
}

// One thread per (edge, 16-byte chunk): gather float4 from source row, scatter-add
// into destination row. AGG buffer (<=102MB) fits in the 192MB L2, so these
// atomics mostly resolve in L2.
__global__ void aggregate_kernel(const float* __restrict__ X,
                                 const long long* __restrict__ src,
                                 const long long* __restrict__ dst,
                                 float* __restrict__ AGG,
                                 int E, int Fin, int chunkShift) {
    long long tid = (long long)blockIdx.x * blockDim.x + threadIdx.x;
    long long total = (long long)E << chunkShift;
    if (tid >= total) return;
    int e = (int)(tid >> chunkShift);
    int c = (int)(tid & ((1 << chunkShift) - 1));
    long long s = src[e];
    long long d = dst[e];
    float4 v = reinterpret_cast<const float4*>(X + s * (long long)Fin)[c];
    float* out = AGG + d * (long long)Fin + (long long)c * 4;
    __hip_atomic_fetch_add(&out[0], v.x, __ATOMIC_RELAXED, __HIP_MEMORY_SCOPE_AGENT);
    __hip_atomic_fetch_add(&out[1], v.y, __ATOMIC_RELAXED, __HIP_MEMORY_SCOPE_AGENT);
    __hip_atomic_fetch_add(&out[2], v.z, __ATOMIC_RELAXED, __HIP_MEMORY_SCOPE_AGENT);
    __hip_atomic_fetch_add(&out[3], v.w, __ATOMIC_RELAXED, __HIP_MEMORY_SCOPE_AGENT);
}

// Fused SAGEConv GEMM: out = (AGG .* inv) @ Wl + X @ Wr + b  [+ReLU]
// One wave32 per 16x16 output tile, FP32 WMMA (V_WMMA_F32_16X16X4_F32), K step 4.
// A 16x4 f32 layout: lanes 0-15 hold M=0..15; VGPR0/1 = K=(2*half)+{0,1}.
// B 4x16 f32 layout: mirrored (lanes hold N, halves hold K pairs).
// C/D v8f: row = m0 + i + 8*half, col = c0 + (lane&15).
__global__ void sage_gemm_wmma(const float* __restrict__ X,
                               const float* __restrict__ AGG,
                               const float* __restrict__ inv,
                               const float* __restrict__ Wl,
                               const float* __restrict__ Wr,
                               const float* __restrict__ bias,
                               float* __restrict__ out,
                               int Nrows, int Fin, int Fout, int doRelu) {
    int wave = (blockIdx.x * blockDim.x + threadIdx.x) >> 5;
    int mt = Nrows >> 4;
    int nt = Fout >> 4;
    if (wave >= mt * nt) return;             // wave-uniform exit: EXEC stays all-1s
    int m0 = (wave % mt) << 4;
    int c0 = (wave / mt) << 4;
    int lane = threadIdx.x & 31;
    int half = lane >> 4;
    int r = lane & 15;
    int rowA = m0 + r;
    int col = c0 + r;
    int kb = 2 * half;

    v8f acc = {0.f, 0.f, 0.f, 0.f, 0.f, 0.f, 0.f, 0.f};
    float invr = inv[rowA];
    const float* arow = AGG + (long long)rowA * Fin;
    const float* xrow = X + (long long)rowA * Fin;

    // mean @ Wl
    for (int k0 = 0; k0 < Fin; k0 += 4) {
        v2f a, b;
        a.x = arow[k0 + kb] * invr;
        a.y = arow[k0 + kb + 1] * invr;
        b.x = Wl[(long long)(k0 + kb) * Fout + col];
        b.y = Wl[(long long)(k0 + kb + 1) * Fout + col];
        acc = __builtin_amdgcn_wmma_f32_16x16x4_f32(false, a, false, b,
                                                    (short)0, acc, false, false);
    }
    // x @ Wr
    for (int k0 = 0; k0 < Fin; k0 += 4) {
        v2f a, b;
        a.x = xrow[k0 + kb];
        a.y = xrow[k0 + kb + 1];
        b.x = Wr[(long long)(k0 + kb) * Fout + col];
        b.y = Wr[(long long)(k0 + kb + 1) * Fout + col];
        acc = __builtin_amdgcn_wmma_f32_16x16x4_f32(false, a, false, b,
                                                    (short)0, acc, false, false);
    }

    float bb = bias[col];
    #pragma unroll
    for (int i = 0; i < 8; ++i) {
        float v = acc[i] + bb;
        if (doRelu) v = fmaxf(v, 0.0f);
        out[(long long)(m0 + i + 8 * half) * Fout + col] = v;
    }
}

// One wave32 per row of 64: lane handles cols {lane, lane+32}; shfl_xor reductions.
__global__ void log_softmax64_kernel(const float* __restrict__ in,
                                     float* __restrict__ out, int Nrows) {
    int wave = (blockIdx.x * blockDim.x + threadIdx.x) >> 5;
    if (wave >= Nrows) return;
    int lane = threadIdx.x & 31;
    const float* row = in + (long long)wave * 64;
    float v0 = row[lane];
    float v1 = row[lane + 32];
    float m = fmaxf(v0, v1);
    #pragma unroll
    for (int off = 16; off > 0; off >>= 1)
        m = fmaxf(m, __shfl_xor(m, off, 32));
    float s = expf(v0 - m) + expf(v1 - m);
    #pragma unroll
    for (int off = 16; off > 0; off >>= 1)
        s += __shfl_xor(s, off, 32);
    float ls = logf(s) + m;
    out[(long long)wave * 64 + lane] = v0 - ls;
    out[(long long)wave * 64 + lane + 32] = v1 - ls;
}

static inline size_t align_up(size_t x, size_t a) { return (x + a - 1) & ~(a - 1); }

extern "C" void kernel_launch(void* const* d_in, const int* in_sizes, int n_in,
                              void* d_out, int out_size, void* d_ws, size_t ws_size,
                              hipStream_t stream) {
    const int N = GN, E = GE;
    const float* x = (const float*)d_in[0];
    const long long* ei = (const long long*)d_in[1];   // int64 [2, E]
    const long long* src = ei;
    const long long* dst = ei + E;
    const float* Wl1 = (const float*)d_in[2];
    const float* Wr1 = (const float*)d_in[3];
    const float* b1  = (const float*)d_in[4];
    const float* Wl2 = (const float*)d_in[5];
    const float* Wr2 = (const float*)d_in[6];
    const float* b2  = (const float*)d_in[7];
    const float* Wl3 = (const float*)d_in[8];
    const float* Wr3 = (const float*)d_in[9];
    const float* b3  = (const float*)d_in[10];
    const float* Wl4 = (const float*)d_in[11];
    const float* Wr4 = (const float*)d_in[12];
    const float* b4  = (const float*)d_in[13];
    float* outp = (float*)d_out;

    // workspace carve-up
    char* w = (char*)d_ws;
    size_t o = 0;
    float* inv = (float*)(w + o); o = align_up(o + (size_t)N * 4, 256);
    float* AGG = (float*)(w + o); o = align_up(o + (size_t)N * 256 * 4, 256);
    float* B1  = (float*)(w + o); o = align_up(o + (size_t)N * 256 * 4, 256);
    float* B2  = (float*)(w + o); o = align_up(o + (size_t)N * 256 * 4, 256);
    (void)ws_size; (void)n_in; (void)in_sizes; (void)out_size;

    const int TB = 256;

    // degree counts -> inverse (computed every call; deterministic)
    fill_zero_kernel<<<(N + TB - 1) / TB, TB, 0, stream>>>(inv, N);
    count_kernel<<<(E + TB - 1) / TB, TB, 0, stream>>>(dst, inv, E);
    inv_kernel<<<(N + TB - 1) / TB, TB, 0, stream>>>(inv, N);

    auto run_layer = [&](const float* Xl, int Fin, int Fout,
                         const float* Wl, const float* Wr, const float* bb,
                         float* Yl, int relu) {
        long long aggN = (long long)N * Fin;
        fill_zero_kernel<<<(unsigned)((aggN + TB - 1) / TB), TB, 0, stream>>>(AGG, aggN);
        int shift = (Fin == 256) ? 6 : (Fin == 128) ? 5 : 4;   // Fin/4 chunks per edge
        long long tot = (long long)E << shift;
        aggregate_kernel<<<(unsigned)((tot + TB - 1) / TB), TB, 0, stream>>>(
            Xl, src, dst, AGG, E, Fin, shift);
        long long waves = (long long)(N / 16) * (Fout / 16);
        long long thr = waves * 32;
        sage_gemm_wmma<<<(unsigned)((thr + TB - 1) / TB), TB, 0, stream>>>(
            Xl, AGG, inv, Wl, Wr, bb, Yl, N, Fin, Fout, relu);
    };

    // h1 = relu(sage1(x))            : 128 -> 256
    run_layer(x,  128, 256, Wl1, Wr1, b1, B1, 1);
    // h2 = relu(sage2(h1))           : 256 -> 256  (model-level ReLU fused)
    run_layer(B1, 256, 256, Wl2, Wr2, b2, B2, 1);
    // h3 = relu(sage3(h2))           : 256 -> 64
    run_layer(B2, 256, 64,  Wl3, Wr3, b3, B1, 1);
    // h4 = sage4(h3)                 : 64 -> 64
    run_layer(B1, 64,  64,  Wl4, Wr4, b4, B2, 0);

    // log_softmax over 64 classes
    long long lsThreads = (long long)N * 32;
    log_softmax64_kernel<<<(unsigned)((lsThreads + TB - 1) / TB), TB, 0, stream>>>(
        B2, outp, N);
}